// ResidualGNNLayer_28630251995627
// MI455X (gfx1250) — compile-verified
//
#include <hip/hip_runtime.h>
#include <hip/hip_bf16.h>
#include <math.h>

// Problem constants (fixed by the reference)
#define NN 50000
#define EE 800000
#define DD 128
#define HH 4
#define ND 512           // H*D
#define NH 200000        // N*H

typedef __attribute__((ext_vector_type(2))) float v2f;
typedef __attribute__((ext_vector_type(8))) float v8f;

// ---------------------------------------------------------------------------
// Kernel 0: init scratch (agg=0, denom=0, m=-inf). Deterministic every call.
// ---------------------------------------------------------------------------
__global__ void init_kernel(float* __restrict__ agg, float* __restrict__ mmax,
                            float* __restrict__ denom) {
    int i = blockIdx.x * blockDim.x + threadIdx.x;
    if (i < NN * ND) agg[i] = 0.0f;
    if (i < NH) { mmax[i] = -__builtin_inff(); denom[i] = 0.0f; }
}

// ---------------------------------------------------------------------------
// Kernel 1: xw = x @ W  via V_WMMA_F32_16X16X4_F32 (full f32 precision).
// Block = 256 threads (8 waves). Block tile: 16 (M) x 128 (N). Wave: 16x16.
// A tile staged in LDS, padded row stride 132 floats -> conflict-free frags.
// ---------------------------------------------------------------------------
__global__ void __launch_bounds__(256)
gemm_wmma_kernel(const float* __restrict__ x, const float* __restrict__ W,
                 float* __restrict__ xw) {
    __shared__ float ldsA[16 * 132];

    const int tid  = threadIdx.x;
    const int lane = tid & 31;
    const int wave = tid >> 5;

    const int row0 = blockIdx.x * 16;              // 3125 blocks in M
    const int col0 = blockIdx.y * 128 + wave * 16; // 4 slabs of 128 in N

    // Cooperative load of A tile: 16 rows x 128 cols (contiguous in x).
    const float* xsrc = x + (size_t)row0 * DD;
    for (int i = tid; i < 16 * DD; i += 256) {
        int r = i >> 7, c = i & 127;
        ldsA[r * 132 + c] = xsrc[i];
    }
    __syncthreads();

    v8f acc = {};
    const int r  = lane & 15;          // A row  (M) / B col (N) select
    const int kp = (lane >> 4) * 2;    // K-pair select for this half-wave

    #pragma unroll 4
    for (int k0 = 0; k0 < DD; k0 += 4) {
        // A fragment: 16x4 f32, lane holds A[r][k0+kp], A[r][k0+kp+1]
        v2f a = *(const v2f*)&ldsA[r * 132 + k0 + kp];
        // B fragment: 4x16 f32, lane holds W[k0+kp][col], W[k0+kp+1][col]
        const float* wp = W + (size_t)(k0 + kp) * ND + col0 + r;
        v2f b;
        b.x = wp[0];
        b.y = wp[ND];
        __builtin_prefetch(wp + 4 * ND, 0, 3);   // global_prefetch_b8 of next K step
        acc = __builtin_amdgcn_wmma_f32_16x16x4_f32(
            /*neg_a=*/false, a, /*neg_b=*/false, b,
            /*c_mod=*/(short)0, acc, /*reuse_a=*/false, /*reuse_b=*/false);
    }

    // D layout: VGPR i, lanes 0-15 -> row i, lanes 16-31 -> row i+8.
    const int rowBase = row0 + ((lane >> 4) << 3);
    const int col = col0 + (lane & 15);
    #pragma unroll
    for (int i = 0; i < 8; ++i) {
        xw[(size_t)(rowBase + i) * ND + col] = acc[i];
    }
}

// ---------------------------------------------------------------------------
// Kernel 2: attention coefficients a_src[n,h], a_dst[n,h]. One wave per (n,h).
// ---------------------------------------------------------------------------
__global__ void __launch_bounds__(256)
attn_coef_kernel(const float* __restrict__ xw, const float* __restrict__ att_src,
                 const float* __restrict__ att_dst,
                 float* __restrict__ a_src, float* __restrict__ a_dst) {
    const int wid  = blockIdx.x * 8 + (threadIdx.x >> 5);
    const int lane = threadIdx.x & 31;
    if (wid >= NH) return;
    const int n = wid >> 2, h = wid & 3;

    const float* xp = xw + (size_t)n * ND + h * DD;
    const float* as = att_src + h * DD;
    const float* ad = att_dst + h * DD;

    float s1 = 0.0f, s2 = 0.0f;
    #pragma unroll
    for (int j = 0; j < 4; ++j) {
        int d = lane + 32 * j;
        float v = xp[d];
        s1 += v * as[d];
        s2 += v * ad[d];
    }
    #pragma unroll
    for (int m = 16; m; m >>= 1) {
        s1 += __shfl_xor(s1, m);
        s2 += __shfl_xor(s2, m);
    }
    if (lane == 0) {
        a_src[n * HH + h] = s1;
        a_dst[n * HH + h] = s2;
    }
}

// Sign-aware float atomic max (m initialized to -inf).
__device__ inline void atomicMaxF(float* addr, float val) {
    if (val >= 0.0f) atomicMax((int*)addr, __float_as_int(val));
    else             atomicMin((unsigned int*)addr, __float_as_uint(val));
}

// ---------------------------------------------------------------------------
// Kernel 3: alpha = leaky_relu(a_src[src]+a_dst[dst]); segment max into m.
// One thread per (edge, head): t = e*4 + h.
// ---------------------------------------------------------------------------
__global__ void edge_alpha_max_kernel(const int* __restrict__ ei,
                                      const float* __restrict__ a_src,
                                      const float* __restrict__ a_dst,
                                      float* __restrict__ alpha,
                                      float* __restrict__ mmax) {
    int t = blockIdx.x * blockDim.x + threadIdx.x;
    if (t >= EE * HH) return;
    int e = t >> 2, h = t & 3;
    int s = ei[e], d = ei[EE + e];
    float a = a_src[s * HH + h] + a_dst[d * HH + h];
    a = (a > 0.0f) ? a : 0.2f * a;          // leaky_relu(0.2)
    alpha[t] = a;
    atomicMaxF(&mmax[d * HH + h], a);
}

// ---------------------------------------------------------------------------
// Kernel 4: ex = exp(alpha - m[dst]); denom[dst] += ex (atomic). ex in place.
// ---------------------------------------------------------------------------
__global__ void edge_exp_sum_kernel(const int* __restrict__ ei,
                                    float* __restrict__ alpha,
                                    const float* __restrict__ mmax,
                                    float* __restrict__ denom) {
    int t = blockIdx.x * blockDim.x + threadIdx.x;
    if (t >= EE * HH) return;
    int e = t >> 2, h = t & 3;
    int d = ei[EE + e];
    float ex = __expf(alpha[t] - mmax[d * HH + h]);
    alpha[t] = ex;
    atomicAdd(&denom[d * HH + h], ex);
}

// ---------------------------------------------------------------------------
// Kernel 5: message scatter. One wave per edge; float4 gathers of xw[src],
// f32 atomic adds into agg[dst]. h == rep because 128/4 lanes x float4 = head.
// ---------------------------------------------------------------------------
__global__ void __launch_bounds__(256)
edge_message_kernel(const int* __restrict__ ei, const float* __restrict__ xw,
                    const float* __restrict__ alpha, const float* __restrict__ denom,
                    float* __restrict__ agg) {
    const int e = blockIdx.x * 8 + (threadIdx.x >> 5);
    const int lane = threadIdx.x & 31;
    if (e >= EE) return;
    const int s = ei[e], d = ei[EE + e];

    const float4* xs = (const float4*)(xw + (size_t)s * ND);
    float* ag = agg + (size_t)d * ND;

    #pragma unroll
    for (int h = 0; h < HH; ++h) {
        float att = alpha[e * HH + h] / denom[d * HH + h];
        float4 v = xs[h * 32 + lane];
        int base = h * DD + lane * 4;
        atomicAdd(&ag[base + 0], v.x * att);
        atomicAdd(&ag[base + 1], v.y * att);
        atomicAdd(&ag[base + 2], v.z * att);
        atomicAdd(&ag[base + 3], v.w * att);
    }
}

// ---------------------------------------------------------------------------
// Kernel 6: mean heads + bias, LayerNorm, exact GELU, residual. Block=node.
// ---------------------------------------------------------------------------
__global__ void __launch_bounds__(128)
finalize_kernel(const float* __restrict__ agg, const float* __restrict__ x,
                const float* __restrict__ bias, const float* __restrict__ gamma,
                const float* __restrict__ beta, float* __restrict__ out) {
    __shared__ float red[8];
    const int n = blockIdx.x;
    const int dlane = threadIdx.x;            // 0..127
    const float* ap = agg + (size_t)n * ND;

    float v = 0.25f * (ap[dlane] + ap[dlane + 128] + ap[dlane + 256] + ap[dlane + 384])
              + bias[dlane];

    float s1 = v, s2 = v * v;
    #pragma unroll
    for (int m = 16; m; m >>= 1) {
        s1 += __shfl_xor(s1, m);
        s2 += __shfl_xor(s2, m);
    }
    int wid = dlane >> 5;
    if ((dlane & 31) == 0) { red[wid] = s1; red[4 + wid] = s2; }
    __syncthreads();
    s1 = red[0] + red[1] + red[2] + red[3];
    s2 = red[4] + red[5] + red[6] + red[7];

    float mu  = s1 * (1.0f / 128.0f);
    float var = s2 * (1.0f / 128.0f) - mu * mu;
    float y = (v - mu) * rsqrtf(var + 1e-5f) * gamma[dlane] + beta[dlane];
    // exact GELU
    y = 0.5f * y * (1.0f + erff(y * 0.70710678118654752f));
    out[(size_t)n * DD + dlane] = y + x[(size_t)n * DD + dlane];
}

// ---------------------------------------------------------------------------
extern "C" void kernel_launch(void* const* d_in, const int* in_sizes, int n_in,
                              void* d_out, int out_size, void* d_ws, size_t ws_size,
                              hipStream_t stream) {
    const float* x       = (const float*)d_in[0];
    const int*   ei      = (const int*)  d_in[1];
    const float* W       = (const float*)d_in[2];
    const float* att_src = (const float*)d_in[3];
    const float* att_dst = (const float*)d_in[4];
    const float* bias    = (const float*)d_in[5];
    const float* gamma   = (const float*)d_in[6];
    const float* beta    = (const float*)d_in[7];
    float* out = (float*)d_out;

    // Workspace layout (floats): xw | agg | a_src | a_dst | m | denom | alpha
    float* ws    = (float*)d_ws;
    float* xw    = ws;
    float* agg   = xw    + (size_t)NN * ND;    // 25.6M
    float* a_src = agg   + (size_t)NN * ND;    // 25.6M
    float* a_dst = a_src + NH;
    float* mmax  = a_dst + NH;
    float* denom = mmax  + NH;
    float* alpha = denom + NH;                 // E*H = 3.2M

    // 0) init accumulators
    init_kernel<<<(NN * ND + 255) / 256, 256, 0, stream>>>(agg, mmax, denom);

    // 1) GEMM via f32 WMMA
    gemm_wmma_kernel<<<dim3(NN / 16, ND / 128), 256, 0, stream>>>(x, W, xw);

    // 2) attention coefficients
    attn_coef_kernel<<<(NH + 7) / 8, 256, 0, stream>>>(xw, att_src, att_dst,
                                                       a_src, a_dst);

    // 3) per-edge alpha + segment max
    edge_alpha_max_kernel<<<(EE * HH + 255) / 256, 256, 0, stream>>>(
        ei, a_src, a_dst, alpha, mmax);

    // 4) per-edge exp + denom
    edge_exp_sum_kernel<<<(EE * HH + 255) / 256, 256, 0, stream>>>(
        ei, alpha, mmax, denom);

    // 5) message scatter (wave per edge)
    edge_message_kernel<<<(EE + 7) / 8, 256, 0, stream>>>(
        ei, xw, alpha, denom, agg);

    // 6) mean heads + LN + GELU + residual
    finalize_kernel<<<NN, 128, 0, stream>>>(agg, x, bias, gamma, beta, out);
}